// IntraAttention_Encoder_48902497632705
// MI455X (gfx1250) — compile-verified
//
#include <hip/hip_runtime.h>

// MI455X / gfx1250, wave32. Exact-f32 WMMA path (V_WMMA_F32_16X16X4_F32):
// workload is HBM-bound (~270MB @ 23.3TB/s ~= 11.6us >> matrix-core time),
// so we keep full f32 precision and still run on the WMMA pipes.
// Staging uses CDNA5 GLOBAL_LOAD_ASYNC_TO_LDS_B128 with double-buffered LDS
// tiles so tile fill (ASYNCcnt) overlaps the WMMA stream.

typedef float v2f __attribute__((ext_vector_type(2)));
typedef float v8f __attribute__((ext_vector_type(8)));

#define B_   16
#define S_   2048
#define H_   1024
#define BM   128          // rows per workgroup (8 waves x 16-row m-tile each)
#define BK   32           // K chunk staged in LDS
#define BKP  36           // padded pitch (144B row stride -> no 16-way bank conflicts)
#define EPSF 1e-12f

// ---- CDNA5 async global->LDS copy (per-lane 16B), tracked by ASYNCcnt ----
__device__ __forceinline__ void async_ld_b128(void* lds_dst, const void* gsrc) {
  // Low 32 bits of a generic shared pointer == LDS byte offset (aperture base
  // has zero low bits; HW truncates to addr[31:0] for LDS anyway).
  unsigned lds_off = (unsigned)(unsigned long long)lds_dst;
  asm volatile("global_load_async_to_lds_b128 %0, %1, off"
               :: "v"(lds_off), "v"(gsrc)
               : "memory");
}
__device__ __forceinline__ void wait_async0() {
  asm volatile("s_wait_asynccnt 0x0" ::: "memory");
}

// ---------------- Kernel 1: dec_fea = st_hat @ W_s^T + W_s_b ----------------
__global__ __launch_bounds__(256)
void dec_fea_kernel(const float* __restrict__ st_hat,
                    const float* __restrict__ W_s,
                    const float* __restrict__ W_sb,
                    float* __restrict__ dec) {
  int id = blockIdx.x * 256 + threadIdx.x;   // [0, 16384)
  int b = id >> 10, n = id & 1023;
  const float4* wr = (const float4*)(W_s + (size_t)n * H_);
  const float4* sr = (const float4*)(st_hat + (size_t)b * H_);
  float acc = W_sb[n];
#pragma unroll 4
  for (int i = 0; i < H_ / 4; ++i) {
    float4 w4 = wr[i], s4 = sr[i];
    acc += w4.x * s4.x + w4.y * s4.y + w4.z * s4.z + w4.w * s4.w;
  }
  dec[id] = acc;
}

// ---- Kernel 2: fused  scores = exp(tanh(h@W_h^T + dec) . v) -> at_un, sum_new ----
// grid = M/BM = 256 blocks, 256 threads (8 wave32). Each wave owns one 16-row
// m-tile; per n-pass all waves share double-buffered LDS tiles of h and W_h and
// each wave keeps 8 WMMA f32 accumulators (16 rows x 128 cols of C in regs).
__global__ __launch_bounds__(256)
void scores_kernel(const float* __restrict__ h,
                   const float* __restrict__ W_h,
                   const float* __restrict__ dec,
                   const float* __restrict__ sumt,
                   const float* __restrict__ mask,
                   const float* __restrict__ vw,
                   float* __restrict__ at_un,     // ws, [B*S]
                   float* __restrict__ bsums,     // ws, [256] per-block mask-weight sums
                   float* __restrict__ sum_new) { // d_out section 3, [B*S]
  __shared__ float hs[2][BM][BKP];
  __shared__ float Ws[2][BM][BKP];
  __shared__ float red[16];

  const int t     = threadIdx.x;
  const int w     = t >> 5;          // wave id 0..7
  const int lane  = t & 31;
  const int mlane = lane & 15;       // N-lane (B/C) or M-lane (A)
  const int koff  = (lane >> 4) * 2; // K sub-offset per CDNA5 ISA A/B layout
  const int m0    = blockIdx.x * BM;
  const int b     = m0 / S_;

  // this thread's 4 staging slots (row, float4-col), fixed across chunks
  const int sr0 = t >> 3, sc0 = (t & 7) * 4;   // + i*32 rows, i = 0..3

  float sacc[8] = {0.f,0.f,0.f,0.f,0.f,0.f,0.f,0.f}; // per-lane v-dot partials

  for (int p = 0; p < 8; ++p) {        // 8 n-passes x 128 cols = N=1024
    const int nb = p * 128;
    v8f C[8] = {};                     // 8 C tiles: cols [nb, nb+128)

    // prologue: async-fill buffer 0 with chunk k0 = 0
#pragma unroll
    for (int i = 0; i < 4; ++i) {
      int r = sr0 + i * 32;
      async_ld_b128(&hs[0][r][sc0], h   + (size_t)(m0 + r) * H_ + sc0);
      async_ld_b128(&Ws[0][r][sc0], W_h + (size_t)(nb + r) * H_ + sc0);
    }

    for (int k0 = 0; k0 < H_; k0 += BK) {
      const int cur = (k0 >> 5) & 1;
      wait_async0();                   // own fills done...
      __syncthreads();                 // ...and everyone's, before reading
      if (k0 + BK < H_) {              // async-prefetch next chunk into other buf
#pragma unroll
        for (int i = 0; i < 4; ++i) {
          int r = sr0 + i * 32;
          async_ld_b128(&hs[cur ^ 1][r][sc0],
                        h   + (size_t)(m0 + r) * H_ + (k0 + BK) + sc0);
          async_ld_b128(&Ws[cur ^ 1][r][sc0],
                        W_h + (size_t)(nb + r) * H_ + (k0 + BK) + sc0);
        }
      }
#pragma unroll
      for (int kk = 0; kk < BK; kk += 4) {
        v2f a;                                 // A 16x4 frag (2 VGPRs)
        a.x = hs[cur][w * 16 + mlane][kk + koff];
        a.y = hs[cur][w * 16 + mlane][kk + koff + 1];
#pragma unroll
        for (int j = 0; j < 8; ++j) {
          v2f bb;                              // B 4x16 frag (2 VGPRs)
          bb.x = Ws[cur][j * 16 + mlane][kk + koff];
          bb.y = Ws[cur][j * 16 + mlane][kk + koff + 1];
          C[j] = __builtin_amdgcn_wmma_f32_16x16x4_f32(
              false, a, false, bb, (short)0, C[j], false, false);
        }
      }
      __syncthreads();                 // (forces dscnt 0) reads done before refill
    }
    // fuse tanh(et + dec_fea) * v into per-lane score partials
#pragma unroll
    for (int j = 0; j < 8; ++j) {
      int n = nb + j * 16 + mlane;
      float dv = dec[b * H_ + n];
      float vv = vw[n];
#pragma unroll
      for (int r = 0; r < 8; ++r)
        sacc[r] += tanhf(C[j][r] + dv) * vv;
    }
  }

  // reduce across the 16-lane halves (C layout: VGPR r -> M=r (lo), M=r+8 (hi))
#pragma unroll
  for (int r = 0; r < 8; ++r) {
    float v = sacc[r];
    v += __shfl_xor(v, 1, 32);
    v += __shfl_xor(v, 2, 32);
    v += __shfl_xor(v, 4, 32);
    v += __shfl_xor(v, 8, 32);
    sacc[r] = v;
  }
  if (mlane == 0) {                 // lanes 0 and 16: 8 rows each
    int hi = lane >> 4;
    float local8 = 0.f;
#pragma unroll
    for (int r = 0; r < 8; ++r) {
      int m  = m0 + w * 16 + r + 8 * hi;   // flat (b,s)
      float e  = expf(sacc[r]);
      float st = sumt[m];
      sum_new[m] = st + e;
      float au = (e / st) * mask[m];
      at_un[m] = au;
      local8 += au;
    }
    red[w * 2 + hi] = local8;
  }
  __syncthreads();
  if (t == 0) {                      // deterministic block partial sum
    float sb = 0.f;
#pragma unroll
    for (int i = 0; i < 16; ++i) sb += red[i];
    bsums[blockIdx.x] = sb;
  }
}

// ---- Kernel 3: ct partials over S-chunks (+ write normalized at) ----
// grid 512: bx -> (b, colchunk cc in 0..3, schunk sc in 0..7). Unique writers.
__global__ __launch_bounds__(256)
void ctx_partial_kernel(const float* __restrict__ h,
                        const float* __restrict__ at_un,
                        const float* __restrict__ bsums,
                        float* __restrict__ ctp,      // ws [8][16][1024]
                        float* __restrict__ at_out) { // d_out section 2
  int bx = blockIdx.x;
  int b  = bx >> 5;
  int cc = (bx >> 3) & 3;
  int sc = bx & 7;
  int t  = threadIdx.x;
  int col = cc * 256 + t;

  float sum16 = 0.f;                   // deterministic fixed-order sum
#pragma unroll
  for (int i = 0; i < 16; ++i) sum16 += bsums[b * 16 + i];
  float inv = 1.f / (sum16 + EPSF);

  int s0 = sc * 256;
  const float* hb = h + (size_t)b * S_ * H_;
  float acc = 0.f;
  for (int s = s0; s < s0 + 256; ++s) {
    if (s + 16 < s0 + 256)                       // -> global_prefetch_b8
      __builtin_prefetch(hb + (size_t)(s + 16) * H_ + col, 0, 0);
    float a = at_un[b * S_ + s] * inv;           // uniform -> scalar load
    acc += a * hb[(size_t)s * H_ + col];         // coalesced
  }
  ctp[((size_t)sc * B_ + b) * H_ + col] = acc;
  if (cc == 0)                                    // single writer per at elem
    at_out[b * S_ + s0 + t] = at_un[b * S_ + s0 + t] * inv;
}

// ---- Kernel 4: reduce 8 ct partials -> ct_e (d_out section 1) ----
__global__ __launch_bounds__(256)
void ctx_reduce_kernel(const float* __restrict__ ctp, float* __restrict__ ct) {
  int id = blockIdx.x * 256 + threadIdx.x;  // [0, 16384)
  int b = id >> 10, col = id & 1023;
  float acc = 0.f;
#pragma unroll
  for (int sc = 0; sc < 8; ++sc)
    acc += ctp[((size_t)sc * B_ + b) * H_ + col];
  ct[id] = acc;
}

extern "C" void kernel_launch(void* const* d_in, const int* in_sizes, int n_in,
                              void* d_out, int out_size, void* d_ws, size_t ws_size,
                              hipStream_t stream) {
  (void)in_sizes; (void)n_in; (void)out_size; (void)ws_size;
  const float* st_hat = (const float*)d_in[0];
  const float* h      = (const float*)d_in[1];
  const float* mask   = (const float*)d_in[2];
  const float* sumt   = (const float*)d_in[3];
  const float* W_h    = (const float*)d_in[4];
  const float* W_s    = (const float*)d_in[5];
  const float* W_sb   = (const float*)d_in[6];
  const float* v_w    = (const float*)d_in[7];
  float* out = (float*)d_out;
  float* ws  = (float*)d_ws;

  // ws layout (floats): dec[16384] | at_un[32768] | bsums[256] | ctp[131072]
  float* dec   = ws;
  float* at_un = ws + 16384;
  float* bsums = ws + 49152;
  float* ctp   = ws + 49408;

  // out layout (floats): ct_e[16384] | at[32768] | sum_new[32768]
  float* out_ct  = out;
  float* out_at  = out + 16384;
  float* out_sum = out + 49152;

  dec_fea_kernel    <<< 64, 256, 0, stream>>>(st_hat, W_s, W_sb, dec);
  scores_kernel     <<<256, 256, 0, stream>>>(h, W_h, dec, sumt, mask, v_w,
                                              at_un, bsums, out_sum);
  ctx_partial_kernel<<<512, 256, 0, stream>>>(h, at_un, bsums, ctp, out_at);
  ctx_reduce_kernel <<< 64, 256, 0, stream>>>(ctp, out_ct);
}